// HistLoss_10393820857111
// MI455X (gfx1250) — compile-verified
//
#include <hip/hip_runtime.h>

// HistLoss for MI455X (gfx1250, wave32).
//
// Memory-bound streaming pass (369 MB @ 23.3 TB/s ~= 16 us floor):
//   - each thread softmaxes 21 channels of its pixels in registers,
//     accumulates per-channel probability sums in registers,
//   - per-batch class histogram via LDS integer atomics (deterministic),
//   - per-block reduction of the 256x21 register partials is done on the
//     MATRIX pipe with V_WMMA_F32_16X16X4_F32 (A = ones 16x4, B = 4-thread
//     x 16-channel LDS tile, chained f32 accumulation). With A == ones the
//     result D[m][n] = sum_k B[k][n] regardless of the exact K<->lane
//     striping, so the reduction is layout-robust.
//   - deterministic finalize kernel sums block partials + histogram.

typedef __attribute__((ext_vector_type(2))) float v2f;
typedef __attribute__((ext_vector_type(8))) float v8f;

#define C_CH 21
#define NB 16
#define HW (512 * 512)
#define CHUNKS 64
#define PIX_PER_CHUNK (HW / CHUNKS)            // 4096
#define TPB 256
#define PIX_PER_THREAD (PIX_PER_CHUNK / TPB)   // 16
#define CPAD 32                                 // channels padded to 32 (2 WMMA col groups)
#define TSTRIDE 260                             // LDS row stride (260 % 64 = 4 -> conflict-free tiles)

// d_ws layout:
//   float partial[NB][CHUNKS][CPAD]   at offset 0                (131072 B)
//   uint  counts [NB][CPAD]           at offset PARTIAL_BYTES    (2048 B)
#define PARTIAL_BYTES ((size_t)NB * CHUNKS * CPAD * sizeof(float))

__global__ void zero_counts_kernel(unsigned* __restrict__ counts) {
    int i = blockIdx.x * blockDim.x + threadIdx.x;
    if (i < NB * CPAD) counts[i] = 0u;
}

__global__ __launch_bounds__(TPB)
void hist_softmax_kernel(const float* __restrict__ pred,
                         const int* __restrict__ trg,
                         float* __restrict__ partial,
                         unsigned* __restrict__ counts) {
    __shared__ float smem[CPAD * TSTRIDE];   // smem[c][t] = partial of thread t, channel c
    __shared__ unsigned hist[CPAD];

    const int tid   = threadIdx.x;
    const int chunk = blockIdx.x;
    const int b     = blockIdx.y;

    if (tid < CPAD) hist[tid] = 0u;
    __syncthreads();

    const float* predb = pred + (size_t)b * C_CH * HW;
    const int*   trgb  = trg + (size_t)b * HW;

    float acc[C_CH];
#pragma unroll
    for (int c = 0; c < C_CH; ++c) acc[c] = 0.0f;

    const int base = chunk * PIX_PER_CHUNK + tid;
    for (int i = 0; i < PIX_PER_THREAD; ++i) {
        const int p = base + i * TPB;        // coalesced across lanes

        // histogram (labels in [0,21); mask keeps LDS in-bounds regardless)
        int lbl = trgb[p];
        atomicAdd(&hist[lbl & (CPAD - 1)], 1u);

        // register softmax over 21 channels (channel stride = HW floats)
        float x[C_CH];
        float m = -3.402823466e38f;
#pragma unroll
        for (int c = 0; c < C_CH; ++c) {
            x[c] = predb[(size_t)c * HW + p];
            m = fmaxf(m, x[c]);
        }
        float s = 0.0f;
#pragma unroll
        for (int c = 0; c < C_CH; ++c) {
            x[c] = __expf(x[c] - m);         // v_exp_f32 path
            s += x[c];
        }
        const float inv = __builtin_amdgcn_rcpf(s);
#pragma unroll
        for (int c = 0; c < C_CH; ++c) acc[c] = fmaf(x[c], inv, acc[c]);
    }

    // dump register partials to LDS: smem[c][tid]; zero-pad channels 21..31
#pragma unroll
    for (int c = 0; c < C_CH; ++c) smem[c * TSTRIDE + tid] = acc[c];
#pragma unroll
    for (int c = C_CH; c < CPAD; ++c) smem[c * TSTRIDE + tid] = 0.0f;
    __syncthreads();

    // ---- WMMA block reduction (wave 0, EXEC all-ones within the wave) ----
    // D = ones[16x4] x B[4x16] + D, chained over 64 chunks of 4 threads.
    // Column n of group g accumulates sum over all 256 threads of channel
    // (16*g + n). Every output row is identical (A rows identical).
    if (tid < 32) {
        const int n = tid & 15;
        const int h = tid >> 4;
        v2f a; a[0] = 1.0f; a[1] = 1.0f;
        v8f d0 = {};
        v8f d1 = {};
        for (int t0 = 0; t0 < TPB; t0 += 4) {
            const int row = t0 + 2 * h;      // this lane supplies threads row, row+1
            v2f b0, b1;
            b0[0] = smem[n * TSTRIDE + row];
            b0[1] = smem[n * TSTRIDE + row + 1];
            b1[0] = smem[(16 + n) * TSTRIDE + row];
            b1[1] = smem[(16 + n) * TSTRIDE + row + 1];
            d0 = __builtin_amdgcn_wmma_f32_16x16x4_f32(
                    false, a, false, b0, (short)0, d0, false, false);
            d1 = __builtin_amdgcn_wmma_f32_16x16x4_f32(
                    false, a, false, b1, (short)0, d1, false, false);
        }
        if (tid < 16) {
            // D row 0: lanes 0..15, VGPR0 -> block sum for channel n / 16+n
            float* outp = partial + ((size_t)b * CHUNKS + chunk) * CPAD;
            outp[n]      = d0[0];
            outp[16 + n] = d1[0];
        }
    }

    // flush histogram (integer atomics: deterministic)
    if (tid < CPAD) {
        unsigned hv = hist[tid];
        if (hv) atomicAdd(&counts[b * CPAD + tid], hv);
    }
}

__global__ __launch_bounds__(512)
void finalize_kernel(const float* __restrict__ partial,
                     const unsigned* __restrict__ counts,
                     float* __restrict__ out) {
    __shared__ float red[512];
    const int tid = threadIdx.x;
    float diff = 0.0f;
    if (tid < NB * C_CH) {
        const int b = tid / C_CH;
        const int c = tid % C_CH;
        float s = 0.0f;
        for (int ch = 0; ch < CHUNKS; ++ch)
            s += partial[((size_t)b * CHUNKS + ch) * CPAD + c];
        const float pred_mean = s * (1.0f / (float)HW);
        const float trg_mean  = (float)counts[b * CPAD + c] * (1.0f / (float)HW);
        diff = fabsf(trg_mean - pred_mean);
    }
    red[tid] = diff;
    __syncthreads();
    for (int s = 256; s > 0; s >>= 1) {
        if (tid < s) red[tid] += red[tid + s];
        __syncthreads();
    }
    if (tid == 0) out[0] = red[0] * (1.0f / (float)NB);
}

extern "C" void kernel_launch(void* const* d_in, const int* in_sizes, int n_in,
                              void* d_out, int out_size, void* d_ws, size_t ws_size,
                              hipStream_t stream) {
    const float* pred = (const float*)d_in[0];
    const int*   trg  = (const int*)d_in[1];   // harness delivers integer inputs as int32

    float*    partial = (float*)d_ws;
    unsigned* counts  = (unsigned*)((char*)d_ws + PARTIAL_BYTES);

    zero_counts_kernel<<<1, 512, 0, stream>>>(counts);

    dim3 grid(CHUNKS, NB);
    hist_softmax_kernel<<<grid, TPB, 0, stream>>>(pred, trg, partial, counts);

    finalize_kernel<<<1, 512, 0, stream>>>(partial, counts, (float*)d_out);
}